// MultiHeadAttention_51453708206744
// MI455X (gfx1250) — compile-verified
//
#include <hip/hip_runtime.h>
#include <math.h>

// MHA: B=2, S=2048, H=1024, HEADS=16, D=64. All fp32 (reference fidelity).
// ws layout (floats): Q[2*16*2048*64] | K | V | CTX[2*2048*1024]  => 64 MB total.

typedef __attribute__((ext_vector_type(2))) float v2f;
typedef __attribute__((ext_vector_type(8))) float v8f;
typedef __attribute__((ext_vector_type(4))) unsigned int u32x4;
typedef __attribute__((ext_vector_type(8))) int i32x8;
typedef __attribute__((ext_vector_type(4))) int i32x4;

#define S_LEN   2048
#define HID     1024
#define NHEAD   16
#define HDIM    64
#define MROWS   4096            // B * S
#define QKV_SZ  (2 * NHEAD * S_LEN * HDIM)   // 4194304 floats per buffer

__device__ __forceinline__ v8f wmma_f32(v2f a, v2f b, v8f c) {
  // V_WMMA_F32_16X16X4_F32 : D = A(16x4 f32) * B(4x16 f32) + C(16x16 f32)
  return __builtin_amdgcn_wmma_f32_16x16x4_f32(false, a, false, b, (short)0, c, false, false);
}

// LDS byte offset of a __shared__ object (ptrtoint of the AS3 pointer).
__device__ __forceinline__ unsigned lds_off(const void* p) {
  return (unsigned)(unsigned long long)(__attribute__((address_space(3))) const void*)p;
}

// ---------------------------------------------------------------------------
// Tensor Data Mover: 2D tile load Global -> LDS with hardware row padding.
// Descriptor built per CDNA5 ISA section 8.3/8.4:
//   g0: count=1 | lds_addr | global_addr[56:0] | type=2
//   g1: data_size=4B, pad_enable, pad_interval/amount codes,
//       tensor_dim0/1 == tile_dim0/1 (no OOB), tensor_dim0_stride = row stride.
// pad codes: interval 4 -> pad after every 32 dwords, 5 -> every 64 dwords;
//            amount 3 -> 4 dwords of padding.  (32->36, 64->68 strides)
// This toolchain declares the 6-arg builtin (extra zero-filled i32x8 group).
// ---------------------------------------------------------------------------
__device__ __forceinline__ void tdm_load_2d(const float* g, unsigned lds_byte,
                                            unsigned tile_w, unsigned tile_h,
                                            unsigned row_stride,
                                            unsigned pad_interval_code,
                                            unsigned pad_amount_code) {
  unsigned long long ga = (unsigned long long)g;
  u32x4 g0;
  g0.x = 1u;                                                  // count=1 (valid user D#)
  g0.y = lds_byte;                                            // lds_addr
  g0.z = (unsigned)ga;                                        // global_addr[31:0]
  g0.w = ((unsigned)(ga >> 32) & 0x01FFFFFFu) | 0x80000000u;  // addr[56:32] | type=2
  i32x8 g1;
  g1[0] = (int)((2u << 16) | (1u << 20) |
                (pad_interval_code << 22) | (pad_amount_code << 25));
  g1[1] = (int)((tile_w & 0xFFFFu) << 16);                        // tensor_dim0 lo16
  g1[2] = (int)(((tile_w >> 16) & 0xFFFFu) | ((tile_h & 0xFFFFu) << 16)); // dim0 hi | dim1 lo
  g1[3] = (int)(((tile_h >> 16) & 0xFFFFu) | ((tile_w & 0xFFFFu) << 16)); // dim1 hi | tile_dim0
  g1[4] = (int)(tile_h & 0xFFFFu);                                // tile_dim1, tile_dim2=0
  g1[5] = (int)row_stride;                                        // tensor_dim0_stride lo32
  g1[6] = 0;
  g1[7] = 0;
  i32x4 z4 = {0, 0, 0, 0};
  i32x8 z8 = {0, 0, 0, 0, 0, 0, 0, 0};
  __builtin_amdgcn_tensor_load_to_lds(g0, g1, z4, z4, z8, 0);
}

// ---------------------------------------------------------------------------
// GEMM: out[M=4096, N=1024] = A[4096,1024] @ W[1024,1024]^T + bias
// permute==1: store to [B, head, S, D] (QKV); else row-major [M, N].
// WG: 256 threads (8 waves), tile 128(M) x 64(N); wave tile 32x32 (4 accs).
// K staged 32 at a time via double-buffered TDM (wave 0 drives the DMA).
// ---------------------------------------------------------------------------
__global__ __launch_bounds__(256) void gemm_bias_kernel(
    const float* __restrict__ A, const float* __restrict__ W,
    const float* __restrict__ bias, float* __restrict__ out, int permute) {
  constexpr int GK = HID;
  __shared__ float As[2][128][36];   // TDM-padded stride 36 (8B aligned, no conflicts)
  __shared__ float Bs[2][64][36];

  const int tid  = threadIdx.x;
  const int wave = tid >> 5;
  const int lane = tid & 31;
  const int half = lane >> 4;
  const int lc   = lane & 15;
  const int wm   = wave & 3;        // 4 M-groups of 32 rows
  const int wn   = wave >> 2;       // 2 N-groups of 32 cols

  const int tileM = blockIdx.y * 128;
  const int tileN = blockIdx.x * 64;

  const float* aBase = A + (size_t)tileM * GK;
  const float* wBase = W + (size_t)tileN * GK;

  if (wave == 0) {                  // prologue: chunk 0 into buffer 0
    tdm_load_2d(aBase, lds_off(&As[0][0][0]), 32, 128, GK, 4, 3);
    tdm_load_2d(wBase, lds_off(&Bs[0][0][0]), 32, 64, GK, 4, 3);
  }

  v8f acc[2][2] = {};

  int cur = 0;
  for (int kb = 0; kb < GK; kb += 32, cur ^= 1) {
    if (wave == 0) {
      if (kb + 32 < GK) {           // issue next chunk, retire current pair
        tdm_load_2d(aBase + kb + 32, lds_off(&As[cur ^ 1][0][0]), 32, 128, GK, 4, 3);
        tdm_load_2d(wBase + kb + 32, lds_off(&Bs[cur ^ 1][0][0]), 32, 64, GK, 4, 3);
        __builtin_amdgcn_s_wait_tensorcnt(2);
      } else {
        __builtin_amdgcn_s_wait_tensorcnt(0);
      }
    }
    __syncthreads();                // data ready for everyone

#pragma unroll
    for (int kk = 0; kk < 8; ++kk) {
      const int k0 = kk * 4 + half * 2;
      v2f a0 = *(const v2f*)&As[cur][wm * 32 + lc][k0];
      v2f a1 = *(const v2f*)&As[cur][wm * 32 + 16 + lc][k0];
      v2f b0 = *(const v2f*)&Bs[cur][wn * 32 + lc][k0];
      v2f b1 = *(const v2f*)&Bs[cur][wn * 32 + 16 + lc][k0];
      acc[0][0] = wmma_f32(a0, b0, acc[0][0]);
      acc[0][1] = wmma_f32(a0, b1, acc[0][1]);
      acc[1][0] = wmma_f32(a1, b0, acc[1][0]);
      acc[1][1] = wmma_f32(a1, b1, acc[1][1]);
    }
    __syncthreads();                // all reads done before buffer reuse
  }

#pragma unroll
  for (int ni = 0; ni < 2; ++ni) {
    const int col = tileN + wn * 32 + ni * 16 + lc;
    const float bv = bias[col];
#pragma unroll
    for (int mi = 0; mi < 2; ++mi) {
#pragma unroll
      for (int i = 0; i < 8; ++i) {
        const int row = tileM + wm * 32 + mi * 16 + i + half * 8;
        const float v = acc[mi][ni][i] + bv;
        if (permute) {
          const int b = row >> 11, s = row & 2047;
          const int h = col >> 6, d = col & 63;
          out[((size_t)((b * NHEAD + h) * S_LEN + s)) * HDIM + d] = v;
        } else {
          out[(size_t)row * HID + col] = v;
        }
      }
    }
  }
}

// ---------------------------------------------------------------------------
// Flash attention (fp32 WMMA, online softmax).
// Grid: (S/128, B*HEADS). Block 256 = 8 waves; each wave owns 16 query rows.
// K/V 16x64 tiles double-buffered into LDS via TDM (wave 0 drives the DMA).
// ---------------------------------------------------------------------------
__global__ __launch_bounds__(256) void attention_kernel(
    const float* __restrict__ Q, const float* __restrict__ K,
    const float* __restrict__ V, float* __restrict__ CTX) {
  __shared__ float Ks[2][16][68];   // TDM-padded stride 68
  __shared__ float Vs[2][16][68];
  __shared__ float Ps[8][16][17];   // per-wave P transpose scratch

  const int tid  = threadIdx.x;
  const int wave = tid >> 5;
  const int lane = tid & 31;
  const int half = lane >> 4;
  const int lc   = lane & 15;

  const int bh    = blockIdx.y;                 // 0..31
  const int qRow0 = blockIdx.x * 128 + wave * 16;

  const float* qptr = Q + (size_t)bh * S_LEN * HDIM;
  const float* kptr = K + (size_t)bh * S_LEN * HDIM;
  const float* vptr = V + (size_t)bh * S_LEN * HDIM;

  // Preload + pre-scale this wave's 16x64 Q tile as A fragments.
  const float scale = 0.125f;       // 1/sqrt(64)
  v2f qa[16];
#pragma unroll
  for (int kk = 0; kk < 16; ++kk) {
    const int k0 = kk * 4 + half * 2;
    v2f q = *(const v2f*)(qptr + (size_t)(qRow0 + lc) * HDIM + k0);
    qa[kk] = q * scale;
  }

  v8f o0 = {}, o1 = {}, o2 = {}, o3 = {};
  float m_run[8], l_run[8];
#pragma unroll
  for (int i = 0; i < 8; ++i) { m_run[i] = -1e30f; l_run[i] = 0.f; }

  if (wave == 0) {                  // prologue: key block 0 into buffer 0
    tdm_load_2d(kptr, lds_off(&Ks[0][0][0]), 64, 16, 64, 5, 3);
    tdm_load_2d(vptr, lds_off(&Vs[0][0][0]), 64, 16, 64, 5, 3);
  }

  constexpr int NBLK = S_LEN / 16;  // 128 key blocks
  int cur = 0;
  for (int j = 0; j < NBLK; ++j, cur ^= 1) {
    if (wave == 0) {
      if (j + 1 < NBLK) {
        tdm_load_2d(kptr + (size_t)(j + 1) * 16 * HDIM,
                    lds_off(&Ks[cur ^ 1][0][0]), 64, 16, 64, 5, 3);
        tdm_load_2d(vptr + (size_t)(j + 1) * 16 * HDIM,
                    lds_off(&Vs[cur ^ 1][0][0]), 64, 16, 64, 5, 3);
        __builtin_amdgcn_s_wait_tensorcnt(2);
      } else {
        __builtin_amdgcn_s_wait_tensorcnt(0);
      }
    }
    __syncthreads();

    // S_tile(16x16) = Qtile(16x64) @ Ktile^T(64x16)  (pre-scaled)
    v8f s = {};
#pragma unroll
    for (int kk = 0; kk < 16; ++kk) {
      const int k0 = kk * 4 + half * 2;
      v2f b = *(const v2f*)&Ks[cur][lc][k0];   // B[k=d][n=key]
      s = wmma_f32(qa[kk], b, s);
    }

    // Online softmax per row (row i+8*half in vgpr i, cols across 16 lanes).
#pragma unroll
    for (int i = 0; i < 8; ++i) {
      float rmax = s[i];
      for (int off = 8; off; off >>= 1)
        rmax = fmaxf(rmax, __shfl_xor(rmax, off, 16));
      const float mnew = fmaxf(m_run[i], rmax);
      const float corr = __expf(m_run[i] - mnew);
      m_run[i] = mnew;
      const float p = __expf(s[i] - mnew);
      s[i] = p;
      float rsum = p;
      for (int off = 8; off; off >>= 1)
        rsum += __shfl_xor(rsum, off, 16);
      l_run[i] = l_run[i] * corr + rsum;
      o0[i] *= corr; o1[i] *= corr; o2[i] *= corr; o3[i] *= corr;
    }

    // Transpose P (C-layout -> A-layout) through per-wave LDS scratch.
#pragma unroll
    for (int i = 0; i < 8; ++i)
      Ps[wave][i + half * 8][lc] = s[i];

    // ctx(16x64) += P(16x16) @ Vtile(16x64)
#pragma unroll
    for (int kk = 0; kk < 4; ++kk) {
      const int k0 = kk * 4 + half * 2;
      v2f a;
      a.x = Ps[wave][lc][k0];
      a.y = Ps[wave][lc][k0 + 1];
      v2f b;
      b.x = Vs[cur][k0][lc];        b.y = Vs[cur][k0 + 1][lc];        o0 = wmma_f32(a, b, o0);
      b.x = Vs[cur][k0][16 + lc];   b.y = Vs[cur][k0 + 1][16 + lc];   o1 = wmma_f32(a, b, o1);
      b.x = Vs[cur][k0][32 + lc];   b.y = Vs[cur][k0 + 1][32 + lc];   o2 = wmma_f32(a, b, o2);
      b.x = Vs[cur][k0][48 + lc];   b.y = Vs[cur][k0 + 1][48 + lc];   o3 = wmma_f32(a, b, o3);
    }
    __syncthreads();                // reads done before buffer reuse
  }

  // Normalize and store ctx to [B, S, 1024] (col = h*64 + d).
  const int b = bh >> 4, h = bh & 15;
#pragma unroll
  for (int i = 0; i < 8; ++i) {
    const float inv = 1.f / l_run[i];
    const int row = qRow0 + i + half * 8;
    float* crow = CTX + ((size_t)(b * S_LEN + row)) * HID + h * HDIM;
    crow[lc]      = o0[i] * inv;
    crow[16 + lc] = o1[i] * inv;
    crow[32 + lc] = o2[i] * inv;
    crow[48 + lc] = o3[i] * inv;
  }
}

// ---------------------------------------------------------------------------
extern "C" void kernel_launch(void* const* d_in, const int* in_sizes, int n_in,
                              void* d_out, int out_size, void* d_ws, size_t ws_size,
                              hipStream_t stream) {
  const float* x  = (const float*)d_in[0];
  const float* Wq = (const float*)d_in[1];
  const float* bq = (const float*)d_in[2];
  const float* Wk = (const float*)d_in[3];
  const float* bk = (const float*)d_in[4];
  const float* Wv = (const float*)d_in[5];
  const float* bv = (const float*)d_in[6];
  const float* Wo = (const float*)d_in[7];
  const float* bo = (const float*)d_in[8];

  float* Qb  = (float*)d_ws;
  float* Kb  = Qb + QKV_SZ;
  float* Vb  = Kb + QKV_SZ;
  float* CTX = Vb + QKV_SZ;

  dim3 gblk(256);
  dim3 ggrid(HID / 64, MROWS / 128);  // (16, 32)

  gemm_bias_kernel<<<ggrid, gblk, 0, stream>>>(x, Wq, bq, Qb, 1);
  gemm_bias_kernel<<<ggrid, gblk, 0, stream>>>(x, Wk, bk, Kb, 1);
  gemm_bias_kernel<<<ggrid, gblk, 0, stream>>>(x, Wv, bv, Vb, 1);

  dim3 agrid(S_LEN / 128, 2 * NHEAD);  // (16, 32)
  attention_kernel<<<agrid, gblk, 0, stream>>>(Qb, Kb, Vb, CTX);

  gemm_bias_kernel<<<ggrid, gblk, 0, stream>>>(CTX, Wo, bo, (float*)d_out, 0);
}